// Polarizability_22308060135981
// MI455X (gfx1250) — compile-verified
//
#include <hip/hip_runtime.h>
#include <hip/hip_bf16.h>
#include <math.h>

// Problem constants (from reference setup_inputs)
#define B_   32
#define A_   512
#define NN_  256
#define NIN_ 256
#define H_   128
#define LOG2F_ 0.69314718055994530942f

#define KSTEPS   (NIN_ / 4)      // 64 k-steps of K=4
#define KSLAB    16              // k-steps staged per LDS slab (32 KB)
#define SLABF    (KSLAB * 8 * 64)  // floats per slab = 8192
#define HSF      (4 * 16 * 132)    // hs bounce region floats = 8448

typedef __attribute__((ext_vector_type(2))) float v2f;
typedef __attribute__((ext_vector_type(8))) float v8f;
typedef __attribute__((ext_vector_type(4))) int   v4i;

// Address-space-qualified vector pointers for the async-to-LDS builtin
typedef __attribute__((address_space(1))) v4i g_v4i;
typedef __attribute__((address_space(3))) v4i l_v4i;

#if defined(__has_builtin)
#if __has_builtin(__builtin_amdgcn_global_load_async_to_lds_b128) && \
    __has_builtin(__builtin_amdgcn_s_wait_asynccnt)
#define USE_ASYNC_LDS 1
#endif
#endif

// ---------------------------------------------------------------------------
// Kernel 0: repack W1 [256,128] into WMMA-fragment order so the GEMM's
// B-operand staging is a pure contiguous stream:
//   W1p[((kstep*8 + t)*32 + lane)*2 + {0,1}] =
//       W1[4*kstep + 2*(lane>=16) + {0,1}][t*16 + (lane&15)]
// ---------------------------------------------------------------------------
__global__ __launch_bounds__(256) void repack_w1_kernel(
    const float* __restrict__ W1, float* __restrict__ W1p)
{
    const int idx = blockIdx.x * 256 + threadIdx.x;  // 0..16383
    const int l   = idx & 31;
    const int t   = (idx >> 5) & 7;
    const int ks  = idx >> 8;                        // 0..63
    const int k   = ks * 4 + (l >> 4) * 2;
    const int col = t * 16 + (l & 15);
    W1p[(size_t)idx * 2 + 0] = W1[(size_t)(k + 0) * H_ + col];
    W1p[(size_t)idx * 2 + 1] = W1[(size_t)(k + 1) * H_ + col];
}

// ---------------------------------------------------------------------------
// Kernel 1: contrib = ssp(rep @ W1 + b1) @ W2 + b2     [B*A, 2]
// 4 waves per block, 16 rows each (64 rows/block). B fragments are staged
// slab-by-slab into LDS with global_load_async_to_lds_b128 (ASYNCcnt) and
// read back with conflict-free ds_load_b64; per k-step all 8 B fragments are
// loaded before the 8 V_WMMA_F32_16X16X4_F32 so the DS latency is paid once.
// ---------------------------------------------------------------------------
__global__ __launch_bounds__(128) void contrib_kernel(
    const float* __restrict__ rep,   // [B*A, NIN_]
    const float* __restrict__ W1p,   // [KSTEPS*8*32*2] fragment-packed
    const float* __restrict__ b1,    // [H_]
    const float* __restrict__ W2,    // [H_, 2]
    const float* __restrict__ b2,    // [2]
    float* __restrict__ contrib)     // [B*A, 2]
{
    __shared__ float smem[HSF];      // max(slab 8192, hs 8448) floats

    const int tid  = threadIdx.x;
    const int wv   = tid >> 5;                 // wave 0..3
    const int lane = tid & 31;
    const int m    = lane & 15;                // M index (A) / N index (B,C,D)
    const int kh   = (lane >> 4) * 2;          // K sub-offset {0,2}
    const int row0 = blockIdx.x * 64 + wv * 16;

    v8f acc[8];
    #pragma unroll
    for (int t = 0; t < 8; ++t) acc[t] = (v8f)0.0f;

    const float* arow = rep + (size_t)(row0 + m) * NIN_;

    for (int s = 0; s < KSTEPS / KSLAB; ++s) {   // 4 slabs of 32 KB
        const float* gsrc = W1p + (size_t)s * SLABF;
#if defined(USE_ASYNC_LDS)
        // CDNA5 async copy: 16 B per lane per issue, ASYNCcnt-tracked
        for (int i = tid; i < SLABF / 4; i += 128) {
            __builtin_amdgcn_global_load_async_to_lds_b128(
                (g_v4i*)(gsrc + (size_t)i * 4),
                (l_v4i*)(smem + (size_t)i * 4),
                0, 0);
        }
        __builtin_amdgcn_s_wait_asynccnt(0);
#else
        for (int i = tid; i < SLABF / 4; i += 128)
            ((float4*)smem)[i] = ((const float4*)gsrc)[i];
#endif
        __syncthreads();

        #pragma unroll 4
        for (int ks = 0; ks < KSLAB; ++ks) {
            const int k = (s * KSLAB + ks) * 4;
            // A fragment (16x4 f32): lane holds rep[row0+m][k+kh+{0,1}]
            v2f a;
            a.x = arow[k + kh + 0];
            a.y = arow[k + kh + 1];
            // Batch all 8 B fragments (4x ds_load_2addr_b64, one wait) ...
            v2f bf[8];
            #pragma unroll
            for (int t = 0; t < 8; ++t)
                bf[t] = *(const v2f*)&smem[(size_t)((ks * 8 + t) * 32 + lane) * 2];
            // ... then the 8 WMMAs issue back-to-back
            #pragma unroll
            for (int t = 0; t < 8; ++t)
                acc[t] = __builtin_amdgcn_wmma_f32_16x16x4_f32(
                    false, a, false, bf[t], (short)0, acc[t], false, false);
        }
        __syncthreads();   // slab consumed by all waves before next overwrite
    }

    // Reuse smem as the per-wave layer-2 bounce buffer.
    // D layout: lane l, VGPR v -> H[row = v + 8*(l>=16)][col = t*16 + (l&15)]
    float* hs = smem + wv * (16 * 132);
    const int mhi = (lane >> 4) * 8;
    #pragma unroll
    for (int t = 0; t < 8; ++t) {
        const int col  = t * 16 + m;
        const float bv = b1[col];
        #pragma unroll
        for (int v = 0; v < 8; ++v) {
            float x  = acc[t][v] + bv;
            // shifted softplus, stable: max(x,0)+log1p(exp(-|x|))-ln2
            float sp = fmaxf(x, 0.0f) + log1pf(expf(-fabsf(x))) - LOG2F_;
            hs[(mhi + v) * 132 + col] = sp;
        }
    }
    __syncthreads();

    // Layer 2: 16 rows x 2 outputs == 32 lanes per wave, one dot(128) each
    const int r = lane & 15;
    const int j = lane >> 4;
    float sum = b2[j];
    #pragma unroll 8
    for (int c = 0; c < H_; ++c) sum += hs[r * 132 + c] * W2[c * 2 + j];
    contrib[(size_t)(row0 + r) * 2 + j] = sum;
}

// ---------------------------------------------------------------------------
// Kernel 2: per-(b,a) pairwise pass. 256 threads = 1 neighbor each.
// Gathers stay L2-resident (10 KB per batch). Reduction: 5-step shfl_xor
// inside each wave32, then one LDS pass across the 8 waves (single barrier).
// ---------------------------------------------------------------------------
__global__ __launch_bounds__(256) void pair_kernel(
    const float* __restrict__ positions, // [B, A, 3]
    const int*   __restrict__ neighbors, // [B, A, NN_]
    const float* __restrict__ nbh_mask,  // [B, A, NN_]
    const float* __restrict__ atom_mask, // [B, A]
    const float* __restrict__ contrib,   // [B, A, 2]
    float* __restrict__ polar_part)      // [B*A, 9]
{
    __shared__ float part[8][6];

    const int tid  = threadIdx.x;    // neighbor index n
    const int wv   = tid >> 5;
    const int lane = tid & 31;
    const int ba   = blockIdx.x;     // b*A + a
    const int b    = ba / A_;

    const size_t pairIdx = (size_t)ba * NN_ + tid;
    const int    j  = neighbors[pairIdx];
    const float  mk = nbh_mask[pairIdx];

    const float cx = positions[(size_t)ba * 3 + 0];
    const float cy = positions[(size_t)ba * 3 + 1];
    const float cz = positions[(size_t)ba * 3 + 2];

    const size_t jp = ((size_t)b * A_ + j) * 3;
    const float dx = positions[jp + 0] - cx;
    const float dy = positions[jp + 1] - cy;
    const float dz = positions[jp + 2] - cz;

    const float dist2 = dx * dx + dy * dy + dz * dz;
    const float dist  = sqrtf(dist2);

    const size_t jc = ((size_t)b * A_ + j) * 2;
    const float nc1 = contrib[jc + 0];
    const float nc2 = contrib[jc + 1];

    // masked_dist = d^3*m + (1-m); field weight = nc2/masked_dist * m
    const float md = dist * dist2 * mk + (1.0f - mk);
    const float w1 = nc1 * mk;
    const float w2 = (nc2 / md) * mk;

    float vals[6] = { dx * w1, dy * w1, dz * w1, dx * w2, dy * w2, dz * w2 };

    #pragma unroll
    for (int off = 16; off > 0; off >>= 1)
        #pragma unroll
        for (int c = 0; c < 6; ++c)
            vals[c] += __shfl_xor(vals[c], off, 32);

    if (lane == 0) {
        #pragma unroll
        for (int c = 0; c < 6; ++c) part[wv][c] = vals[c];
    }
    __syncthreads();

    if (tid == 0) {
        float acc[6] = {0, 0, 0, 0, 0, 0};
        #pragma unroll
        for (int w = 0; w < 8; ++w)
            #pragma unroll
            for (int c = 0; c < 6; ++c) acc[c] += part[w][c];

        float D3[3] = { acc[0], acc[1], acc[2] };
        float F3[3] = { acc[3], acc[4], acc[5] };
        float nrm = sqrtf(F3[0]*F3[0] + F3[1]*F3[1] + F3[2]*F3[2]);
        nrm += (nrm < 1e-10f) ? 1.0f : 0.0f;
        const float inv = 1.0f / nrm;
        F3[0] *= inv; F3[1] *= inv; F3[2] *= inv;
        const float am = atom_mask[ba];
        float* o = polar_part + (size_t)ba * 9;
        #pragma unroll
        for (int r = 0; r < 3; ++r)
            #pragma unroll
            for (int c = 0; c < 3; ++c)
                o[r * 3 + c] = 0.5f * (D3[r] * F3[c] + D3[c] * F3[r]) * am;
    }
}

// ---------------------------------------------------------------------------
// Kernel 3: deterministic per-batch reduction of 512 atom partials -> [B,3,3]
// ---------------------------------------------------------------------------
__global__ __launch_bounds__(256) void reduce_kernel(
    const float* __restrict__ polar_part, // [B*A, 9]
    float* __restrict__ out)              // [B, 9]
{
    __shared__ float red[256][9];
    const int tid = threadIdx.x;
    const int b   = blockIdx.x;

    float acc[9] = {0,0,0,0,0,0,0,0,0};
    for (int a = tid; a < A_; a += 256) {
        const float* p = polar_part + ((size_t)b * A_ + a) * 9;
        #pragma unroll
        for (int k = 0; k < 9; ++k) acc[k] += p[k];
    }
    #pragma unroll
    for (int k = 0; k < 9; ++k) red[tid][k] = acc[k];
    __syncthreads();

    for (int s = 128; s > 0; s >>= 1) {
        if (tid < s) {
            #pragma unroll
            for (int k = 0; k < 9; ++k) red[tid][k] += red[tid + s][k];
        }
        __syncthreads();
    }
    if (tid < 9) out[b * 9 + tid] = red[0][tid];
}

// ---------------------------------------------------------------------------
// Launch: inputs in setup_inputs() order:
// 0 positions, 1 representation, 2 neighbors, 3 nbh_mask, 4 atom_mask,
// 5 W1, 6 b1, 7 W2, 8 b2
// ---------------------------------------------------------------------------
extern "C" void kernel_launch(void* const* d_in, const int* in_sizes, int n_in,
                              void* d_out, int out_size, void* d_ws, size_t ws_size,
                              hipStream_t stream) {
    const float* positions = (const float*)d_in[0];
    const float* rep       = (const float*)d_in[1];
    const int*   neighbors = (const int*)  d_in[2];
    const float* nbh_mask  = (const float*)d_in[3];
    const float* atom_mask = (const float*)d_in[4];
    const float* W1        = (const float*)d_in[5];
    const float* b1        = (const float*)d_in[6];
    const float* W2        = (const float*)d_in[7];
    const float* b2        = (const float*)d_in[8];
    float*       out       = (float*)d_out;

    // Workspace layout (all fully written every call)
    float* contrib    = (float*)d_ws;                          // B*A*2   floats
    float* polar_part = contrib + (size_t)B_ * A_ * 2;         // B*A*9   floats
    float* W1p        = polar_part + (size_t)B_ * A_ * 9;      // 32768   floats

    // 0) repack W1 into WMMA fragment order (one-shot, 128 KB)
    repack_w1_kernel<<<64, 256, 0, stream>>>(W1, W1p);

    // 1) per-atom MLP via f32 WMMA: 256 blocks x 4 waves, 64 rows each
    contrib_kernel<<<(B_ * A_) / 64, 128, 0, stream>>>(rep, W1p, b1, W2, b2, contrib);

    // 2) pairwise pass: one block per atom, one thread per neighbor
    pair_kernel<<<B_ * A_, 256, 0, stream>>>(positions, neighbors, nbh_mask,
                                             atom_mask, contrib, polar_part);

    // 3) per-batch reduction to the 3x3 outputs
    reduce_kernel<<<B_, 256, 0, stream>>>(polar_part, out);
}